// linear_trunc_dep_3762391352095
// MI455X (gfx1250) — compile-verified
//
#include <hip/hip_runtime.h>

#define B_ROWS    65536
#define I_DIM     1024
#define O_DIM     64
#define KTRUNC    32
#define TILE_ROWS 32
#define S_PAD     1028              // padded LDS row stride (floats) -> conflict-free WMMA A loads
#define NWAVES    8
#define NTHREADS  256

typedef __attribute__((ext_vector_type(2))) float v2f;
typedef __attribute__((ext_vector_type(8))) float v8f;

// -------- kernel 1: colsum[i] = sum_o W[o,i] --------------------------------
__global__ void colsum_kernel(const float* __restrict__ W, float* __restrict__ cs) {
    int i = blockIdx.x * blockDim.x + threadIdx.x;
    if (i < I_DIM) {
        float s = 0.0f;
#pragma unroll
        for (int o = 0; o < O_DIM; ++o) s += W[o * I_DIM + i];
        cs[i] = s;
    }
}

// fp32 -> order-preserving u32 key
__device__ __forceinline__ unsigned sortkey(float f) {
    unsigned u = __float_as_uint(f);
    return u ^ (unsigned)(((int)u >> 31) | (int)0x80000000);
}

// load lane's 8 buckets of a 256-entry histogram region
__device__ __forceinline__ void load_cnt(const unsigned* region, int lane, int cnt[8], int& s) {
    uint4 c0 = *(const uint4*)&region[lane * 8 + 0];
    uint4 c1 = *(const uint4*)&region[lane * 8 + 4];
    cnt[0] = (int)c0.x; cnt[1] = (int)c0.y; cnt[2] = (int)c0.z; cnt[3] = (int)c0.w;
    cnt[4] = (int)c1.x; cnt[5] = (int)c1.y; cnt[6] = (int)c1.z; cnt[7] = (int)c1.w;
    s = cnt[0] + cnt[1] + cnt[2] + cnt[3] + cnt[4] + cnt[5] + cnt[6] + cnt[7];
}

// descending scan: bucket containing the `want`-th largest; nwant = rank within bucket
__device__ __forceinline__ void scan_top(const unsigned* region, int lane, int want,
                                         int& bucket, int& nwant) {
    int cnt[8], s;
    load_cnt(region, lane, cnt, s);
    int suf = s;                                   // suffix sum: suf_L = sum_{L' >= L}
#pragma unroll
    for (int off = 1; off < 32; off <<= 1) {
        int tv = __shfl_down(suf, off, 32);
        if (lane + off < 32) suf += tv;
    }
    int cand = (suf >= want) ? lane : 0;           // highest lane still reaching `want`
#pragma unroll
    for (int off = 16; off > 0; off >>= 1) {
        int o = __shfl_xor(cand, off, 32);
        cand = cand > o ? cand : o;
    }
    const int lsel = cand;
    int above = __shfl_down(suf, 1, 32);
    if (lane == 31) above = 0;
    above = __shfl(above, lsel, 32);               // count strictly above lsel's buckets
    int b = 0, nw = want;
    if (lane == lsel) {
        int cum = above, done = 0;
#pragma unroll
        for (int t = 7; t >= 0; --t) {
            int c = cnt[t];
            if (!done) {
                if (cum + c >= want) { b = lane * 8 + t; nw = want - cum; done = 1; }
                else                 { cum += c; }
            }
        }
    }
    bucket = __shfl(b, lsel, 32);
    nwant  = __shfl(nw, lsel, 32);
}

// ascending scan: bucket containing the `want`-th smallest; nwant = rank within bucket
__device__ __forceinline__ void scan_bottom(const unsigned* region, int lane, int want,
                                            int& bucket, int& nwant) {
    int cnt[8], s;
    load_cnt(region, lane, cnt, s);
    int pre = s;                                   // prefix sum: pre_L = sum_{L' <= L}
#pragma unroll
    for (int off = 1; off < 32; off <<= 1) {
        int tv = __shfl_up(pre, off, 32);
        if (lane >= off) pre += tv;
    }
    int cand = (pre >= want) ? lane : 31;          // lowest lane reaching `want`
#pragma unroll
    for (int off = 16; off > 0; off >>= 1) {
        int o = __shfl_xor(cand, off, 32);
        cand = cand < o ? cand : o;
    }
    const int lsel = cand;
    int below = __shfl_up(pre, 1, 32);
    if (lane == 0) below = 0;
    below = __shfl(below, lsel, 32);               // count strictly below lsel's buckets
    int b = 0, nw = want;
    if (lane == lsel) {
        int cum = below, done = 0;
#pragma unroll
        for (int t = 0; t < 8; ++t) {
            int c = cnt[t];
            if (!done) {
                if (cum + c >= want) { b = lane * 8 + t; nw = want - cum; done = 1; }
                else                 { cum += c; }
            }
        }
    }
    bucket = __shfl(b, lsel, 32);
    nwant  = __shfl(nw, lsel, 32);
}

// -------- kernel 2: fused selection + masked GEMM (WMMA f32 16x16x4) --------
__launch_bounds__(NTHREADS)
__global__ void fused_select_gemm(const float* __restrict__ X,
                                  const float* __restrict__ W,
                                  const float* __restrict__ CS,
                                  float* __restrict__ OUT) {
    extern __shared__ char smem_raw[];
    float*    xs   = (float*)smem_raw;                       // TILE_ROWS * S_PAD floats
    unsigned* hist = (unsigned*)(xs + TILE_ROWS * S_PAD);    // NWAVES * 512 u32

    const int tid     = threadIdx.x;
    const int wave    = tid >> 5;
    const int lane    = tid & 31;
    const int rowBase = blockIdx.x * TILE_ROWS;
    unsigned* h = hist + wave * 512;

    // GEMM tile coordinates (also used to prefetch W during selection)
    const int half = lane >> 4;          // 0: K{0,1}, 1: K{2,3}
    const int mm   = lane & 15;
    const int m0   = (wave >> 2) * 16;   // row group 0..1
    const int n0   = (wave & 3) * 16;    // col group 0..3
    const float* wp = W + (size_t)(n0 + mm) * I_DIM + 2 * half;
    __builtin_prefetch(wp, 0, 1);        // global_prefetch: W rows in flight during selection

    // per-lane colsum cache: element i = lane*4 + 128*j  (float4 index lane + 32*j)
    float4 csv[8];
#pragma unroll
    for (int j = 0; j < 8; ++j) csv[j] = ((const float4*)CS)[lane + 32 * j];

    // ---- per-wave: rows wave*4 .. wave*4+3 ----
    for (int rr = 0; rr < 4; ++rr) {
        const int r = wave * 4 + rr;
        const float* xrow = X + (size_t)(rowBase + r) * I_DIM;

        float4  xv[8];
        unsigned keys[32];
#pragma unroll
        for (int j = 0; j < 8; ++j) {
            xv[j] = ((const float4*)xrow)[lane + 32 * j];
            keys[4 * j + 0] = sortkey(xv[j].x * csv[j].x);
            keys[4 * j + 1] = sortkey(xv[j].y * csv[j].y);
            keys[4 * j + 2] = sortkey(xv[j].z * csv[j].z);
            keys[4 * j + 3] = sortkey(xv[j].w * csv[j].w);
        }

        // ---- level 0: one shared 256-bucket histogram serves both tails ----
        uint4 z4 = {0u, 0u, 0u, 0u};
        *(uint4*)&h[lane * 8 + 0] = z4;
        *(uint4*)&h[lane * 8 + 4] = z4;
        __builtin_amdgcn_wave_barrier();
#pragma unroll
        for (int j = 0; j < 32; ++j) atomicAdd(&h[keys[j] >> 24], 1u);
        __builtin_amdgcn_wave_barrier();

        int bhi, want_hi, blo, want_lo;
        scan_top   (h, lane, KTRUNC, bhi, want_hi);
        scan_bottom(h, lane, KTRUNC, blo, want_lo);
        unsigned prehi = (unsigned)bhi << 24;
        unsigned prelo = (unsigned)blo << 24;

        // ---- levels 1..3: dual histograms, one predicated ds_add per element ----
#pragma unroll
        for (int level = 1; level < 4; ++level) {
            const int shift = 24 - level * 8;
            const bool same = (prehi == prelo);   // degenerate: both tails in same bucket chain
            uint4 zz = {0u, 0u, 0u, 0u};
            *(uint4*)&h[lane * 16 +  0] = zz;
            *(uint4*)&h[lane * 16 +  4] = zz;
            *(uint4*)&h[lane * 16 +  8] = zz;
            *(uint4*)&h[lane * 16 + 12] = zz;
            __builtin_amdgcn_wave_barrier();
            const unsigned ph = prehi >> (shift + 8);
            const unsigned pl = prelo >> (shift + 8);
#pragma unroll
            for (int j = 0; j < 32; ++j) {
                const unsigned sh = keys[j] >> shift;
                const unsigned d  = sh & 0xFFu;
                const bool mlo = ((sh >> 8) == pl);
                const bool mhi = ((sh >> 8) == ph);
                if (mhi | mlo) atomicAdd(&h[d + (mlo ? 256u : 0u)], 1u);
            }
            __builtin_amdgcn_wave_barrier();
            scan_top   (same ? h + 256 : h, lane, want_hi, bhi, want_hi);
            scan_bottom(h + 256,            lane, want_lo, blo, want_lo);
            prehi |= (unsigned)bhi << shift;
            prelo |= (unsigned)blo << shift;
        }
        const unsigned Thi = prehi;   // exact key of 32nd largest
        const unsigned Tlo = prelo;   // exact key of 32nd smallest

        // ---- write masked x into LDS in GEMM layout ----
#pragma unroll
        for (int j = 0; j < 8; ++j) {
            float4 v = xv[j];
            const unsigned k0 = keys[4 * j + 0], k1 = keys[4 * j + 1];
            const unsigned k2 = keys[4 * j + 2], k3 = keys[4 * j + 3];
            if (k0 >= Thi || k0 <= Tlo) v.x = 0.0f;
            if (k1 >= Thi || k1 <= Tlo) v.y = 0.0f;
            if (k2 >= Thi || k2 <= Tlo) v.z = 0.0f;
            if (k3 >= Thi || k3 <= Tlo) v.w = 0.0f;
            *(float4*)&xs[r * S_PAD + lane * 4 + 128 * j] = v;
        }
    }
    __syncthreads();

    // ---- GEMM: each wave computes one 16x16 tile of (masked x) @ W^T ----
    const float* ap = xs + (size_t)(m0 + mm) * S_PAD + 2 * half;
    v8f acc = {0.f, 0.f, 0.f, 0.f, 0.f, 0.f, 0.f, 0.f};
#pragma unroll 8
    for (int k0 = 0; k0 < I_DIM; k0 += 4) {
        v2f a = *(const v2f*)(ap + k0);
        v2f b = *(const v2f*)(wp + k0);
        acc = __builtin_amdgcn_wmma_f32_16x16x4_f32(false, a, false, b,
                                                    (short)0, acc, false, false);
    }
    // D layout: VGPR t -> M = t + 8*half, N = mm
    float* op = OUT + (size_t)(rowBase + m0 + half * 8) * O_DIM + n0 + mm;
#pragma unroll
    for (int t = 0; t < 8; ++t) op[(size_t)t * O_DIM] = acc[t];
}

// -------- launcher ----------------------------------------------------------
extern "C" void kernel_launch(void* const* d_in, const int* in_sizes, int n_in,
                              void* d_out, int out_size, void* d_ws, size_t ws_size,
                              hipStream_t stream) {
    const float* x = (const float*)d_in[0];   // (65536, 1024) f32
    const float* w = (const float*)d_in[1];   // (64, 1024) f32
    float* out = (float*)d_out;               // (65536, 64) f32
    float* cs  = (float*)d_ws;                // 1024 f32 scratch

    colsum_kernel<<<I_DIM / 256, 256, 0, stream>>>(w, cs);

    const size_t smem = (size_t)(TILE_ROWS * S_PAD) * 4 + (size_t)(NWAVES * 512) * 4;  // ~144.5 KB
    fused_select_gemm<<<B_ROWS / TILE_ROWS, NTHREADS, smem, stream>>>(x, w, cs, out);
}